// GraphExportAdapter_32504312496171
// MI455X (gfx1250) — compile-verified
//
#include <hip/hip_runtime.h>
#include <hip/hip_bf16.h>
#include <math.h>

typedef __attribute__((ext_vector_type(2))) float v2f;
typedef __attribute__((ext_vector_type(8))) float v8f;

static constexpr int kH  = 512;
static constexpr int kW  = 512;
static constexpr int kHW = kH * kW;          // 2^18
static constexpr int kHW4 = kHW / 4;         // 2^16 float4s per plane
static constexpr int kB  = 8;
static constexpr int kTY = 7;                // tiles per dim
static constexpr int kNT = kTY * kTY;        // 49 tiles
static constexpr int kCellsPerImg = 64;      // 8x8 cells of 64x64

// ---------------------------------------------------------------------------
// Kernel A: per-(batch, 64x64-cell) masked reduction, float4 (b128) loads.
// acc = sum over cell of mask[b,0,y,x] * (img[b,0]+img[b,1]+img[b,2])
// Block reduction via V_WMMA_F32_16X16X4_F32 against an all-ones B matrix:
// four accumulating WMMAs collapse the 256 LDS partials; per-lane sum of the
// 8 D VGPRs gives rows 0-7 (lanes 0-15) / 8-15 (lanes 16-31); one
// shfl_xor(16) finishes.  All waves run it redundantly so EXEC is all ones.
// ---------------------------------------------------------------------------
__global__ __launch_bounds__(256) void cell_sum_kernel(
    const float* __restrict__ image, const float* __restrict__ mask,
    float* __restrict__ cell) {
  __shared__ float red[256];
  const int blk  = blockIdx.x;
  const int b    = blk >> 6;          // 64 cells per image
  const int cidx = blk & 63;
  const int cy = cidx >> 3, cx = cidx & 7;
  const float4* img4 = reinterpret_cast<const float4*>(image) + (size_t)b * 3 * kHW4;
  const float4* m4   = reinterpret_cast<const float4*>(mask)  + (size_t)b * kHW4;
  const int y0 = cy << 6, x0 = cx << 6;

  float acc = 0.0f;
  // 64 rows x 16 float4/row = 1024 float4 per cell -> 4 iters/thread
  for (int i = threadIdx.x; i < 1024; i += 256) {
    const int row = i >> 4, col = i & 15;
    const size_t off = ((size_t)(y0 + row) * kW + x0) / 4 + col;
    const float4 m  = m4[off];
    const float4 c0 = img4[off];
    const float4 c1 = img4[off + kHW4];
    const float4 c2 = img4[off + 2 * kHW4];
    acc = fmaf(m.x, c0.x + c1.x + c2.x, acc);
    acc = fmaf(m.y, c0.y + c1.y + c2.y, acc);
    acc = fmaf(m.z, c0.z + c1.z + c2.z, acc);
    acc = fmaf(m.w, c0.w + c1.w + c2.w, acc);
  }
  red[threadIdx.x] = acc;
  __syncthreads();

  const int lane = threadIdx.x & 31;
  const int row  = lane & 15;
  const int khi  = (lane >> 4) << 1;  // lanes 0-15: K=0,1 ; lanes 16-31: K=2,3
  v2f ones; ones.x = 1.0f; ones.y = 1.0f;
  v8f dacc = {};
#pragma unroll
  for (int ch = 0; ch < 4; ++ch) {    // four 16x4 chunks of the 256 partials
    const float* base = red + ch * 64;
    v2f a;
    a.x = base[row * 4 + khi + 0];
    a.y = base[row * 4 + khi + 1];
    dacc = __builtin_amdgcn_wmma_f32_16x16x4_f32(
        false, a, false, ones, (short)0, dacc, false, false);
  }
  float s = dacc[0] + dacc[1] + dacc[2] + dacc[3] +
            dacc[4] + dacc[5] + dacc[6] + dacc[7];
  float tot = s + __shfl_xor(s, 16, 32);
  if (threadIdx.x == 0) cell[blk] = tot;
}

// ---------------------------------------------------------------------------
// Kernel B: tile logits.  Tile (ty,tx) == 2x2 block of cells
// {ty,ty+1} x {tx,tx+1}.  Mean divisor = C*TILE*TILE = 49152.
// ---------------------------------------------------------------------------
__global__ void logits_kernel(const float* __restrict__ cell,
                              float* __restrict__ logits) {
  const int i = blockIdx.x * blockDim.x + threadIdx.x;
  if (i >= kB * kNT) return;
  const int b  = i / kNT;
  const int tt = i - b * kNT;
  const int ty = tt / kTY, tx = tt - ty * kTY;
  const float* cb = cell + b * kCellsPerImg;
  const float sum = cb[ty * 8 + tx]       + cb[ty * 8 + tx + 1] +
                    cb[(ty + 1) * 8 + tx] + cb[(ty + 1) * 8 + tx + 1];
  logits[i] = sum * (1.0f / 49152.0f);
}

// ---------------------------------------------------------------------------
// Kernel C: elementwise output, 4 pixels/thread (b128 path).
// 4 aligned consecutive pixels share (qy,qx), so the covering-tile sum and
// count are computed once per float4.  Tiles covering (h,w): ty in
// {q-1,q} ∩ [0,6] with q=h>>6 (same for x); count = product of valid extents
// (always >= 1, matching clip(counts,1)).  Batch logits staged in LDS.
// ---------------------------------------------------------------------------
__global__ __launch_bounds__(256) void out_kernel(
    const float* __restrict__ gheat, const float* __restrict__ mask,
    const float* __restrict__ logits, float* __restrict__ out) {
  __shared__ float lg[kNT];
  const size_t i4 = (size_t)blockIdx.x * 256 + threadIdx.x;   // float4 index
  const int b  = (int)(i4 >> 16);          // kHW4 = 2^16; blocks never straddle b
  const int p4 = (int)(i4 & (kHW4 - 1));
  if (threadIdx.x < kNT) lg[threadIdx.x] = logits[b * kNT + threadIdx.x];
  __syncthreads();

  const int h = p4 >> 7;                   // 128 float4 per row
  const int w = (p4 & 127) << 2;
  const int qy = h >> 6, qx = w >> 6;
  int tys[2], ntys = 0;
  if (qy >= 1)       tys[ntys++] = qy - 1;
  if (qy <= kTY - 1) tys[ntys++] = qy;
  int txs[2], ntxs = 0;
  if (qx >= 1)       txs[ntxs++] = qx - 1;
  if (qx <= kTY - 1) txs[ntxs++] = qx;

  float local = 0.0f;
  for (int a = 0; a < ntys; ++a)
    for (int c = 0; c < ntxs; ++c)
      local += lg[tys[a] * kTY + txs[c]];
  const float linv = local / (float)(ntys * ntxs);   // shared by all 4 pixels

  const float4 m = reinterpret_cast<const float4*>(mask)[i4];
  const float4 g = reinterpret_cast<const float4*>(gheat)[i4];
  float4 o;
  {
    const float x = 0.5f * g.x + 0.5f * (linv * m.x);
    o.x = m.x / (1.0f + __expf(-x));
  }
  {
    const float x = 0.5f * g.y + 0.5f * (linv * m.y);
    o.y = m.y / (1.0f + __expf(-x));
  }
  {
    const float x = 0.5f * g.z + 0.5f * (linv * m.z);
    o.z = m.z / (1.0f + __expf(-x));
  }
  {
    const float x = 0.5f * g.w + 0.5f * (linv * m.w);
    o.w = m.w / (1.0f + __expf(-x));
  }
  reinterpret_cast<float4*>(out)[i4] = o;
}

// ---------------------------------------------------------------------------
extern "C" void kernel_launch(void* const* d_in, const int* in_sizes, int n_in,
                              void* d_out, int out_size, void* d_ws,
                              size_t ws_size, hipStream_t stream) {
  const float* image = (const float*)d_in[0];  // (8,3,512,512)
  const float* vmask = (const float*)d_in[1];  // (8,1,512,512)
  const float* gheat = (const float*)d_in[2];  // (8,1,512,512)
  // d_in[3] tile_mask_bank, d_in[4] counts, d_in[5/6] row/col idx: the tiling
  // structure is closed-form, so these are intentionally unused.
  (void)in_sizes; (void)n_in; (void)out_size; (void)ws_size;

  float* out    = (float*)d_out;
  float* ws     = (float*)d_ws;
  float* cell   = ws;            // 512 floats
  float* logits = ws + 512;      // 392 floats

  cell_sum_kernel<<<kB * kCellsPerImg, 256, 0, stream>>>(image, vmask, cell);
  logits_kernel<<<(kB * kNT + 255) / 256, 256, 0, stream>>>(cell, logits);
  out_kernel<<<(kB * kHW4) / 256, 256, 0, stream>>>(gheat, vmask, logits, out);
}